// BVH_38336878084112
// MI455X (gfx1250) — compile-verified
//
#include <hip/hip_runtime.h>

#define TRI_F   1024
#define TILES   64            // TRI_F / 16
#define NTP     2080          // TILES*(TILES+1)/2 upper-tri tile pairs
#define KOUT    8192          // TRI_F * MAX_COLLISIONS
#define EPSF    1e-6f
#define WPB     (TRI_F * TILES)   // mask words per batch = 65536

typedef __attribute__((ext_vector_type(2))) float v2f;
typedef __attribute__((ext_vector_type(8))) float v8f;

#if defined(__AMDGCN__) && __has_builtin(__builtin_amdgcn_global_load_async_to_lds_b32)
#define USE_ASYNC_LDS 1
#else
#define USE_ASYNC_LDS 0
#endif

// ---------------------------------------------------------------------------
// Per-triangle plane: N = cross(v1-v0, v2-v0), d = -dot(N, v0)
// ---------------------------------------------------------------------------
__global__ __launch_bounds__(256) void plane_kernel(const float* __restrict__ tris,
                                                    float* __restrict__ planes,
                                                    int nTri) {
    int i = blockIdx.x * blockDim.x + threadIdx.x;
    if (i >= nTri) return;
    const float* v = tris + (size_t)i * 9;
    float e1x = v[3] - v[0], e1y = v[4] - v[1], e1z = v[5] - v[2];
    float e2x = v[6] - v[0], e2y = v[7] - v[1], e2z = v[8] - v[2];
    float nx = e1y * e2z - e1z * e2y;
    float ny = e1z * e2x - e1x * e2z;
    float nz = e1x * e2y - e1y * e2x;
    float d  = -(nx * v[0] + ny * v[1] + nz * v[2]);
    planes[i * 4 + 0] = nx;
    planes[i * 4 + 1] = ny;
    planes[i * 4 + 2] = nz;
    planes[i * 4 + 3] = d;
}

// ---------------------------------------------------------------------------
// Moller interval helpers (match reference _interval / seg exactly)
// ---------------------------------------------------------------------------
__device__ __forceinline__ void seg_lohi(float pa, float da, float pb, float db,
                                         float pc, float dc, float& lo, float& hi) {
    float den1 = db - da; if (fabsf(den1) < 1e-12f) den1 = 1e-12f;
    float den2 = dc - da; if (fabsf(den2) < 1e-12f) den2 = 1e-12f;
    float t1 = pb + (pa - pb) * db / den1;
    float t2 = pc + (pa - pc) * dc / den2;
    lo = fminf(t1, t2);
    hi = fmaxf(t1, t2);
}

__device__ __forceinline__ void interval(const float p0, const float p1, const float p2,
                                         const float d0, const float d1, const float d2,
                                         float& lo, float& hi) {
    float lo2, hi2, lo1, hi1, lo0, hi0;
    seg_lohi(p2, d2, p0, d0, p1, d1, lo2, hi2);
    seg_lohi(p1, d1, p0, d0, p2, d2, lo1, hi1);
    seg_lohi(p0, d0, p1, d1, p2, d2, lo0, hi0);
    bool c2 = (d0 * d1) > 0.0f;
    bool c1 = (!c2) && ((d0 * d2) > 0.0f);
    lo = c2 ? lo2 : (c1 ? lo1 : lo0);
    hi = c2 ? hi2 : (c1 ? hi1 : hi0);
}

// ---------------------------------------------------------------------------
// Pair kernel: one block = one 16x16 tile of (f,g) pairs, upper triangle.
// Tile data staged global->LDS via GLOBAL_LOAD_ASYNC_TO_LDS (ASYNCcnt),
// WMMA (f32 16x16x4) computes the plane-distance matrices Du_k / Dv_k.
// ---------------------------------------------------------------------------
__global__ __launch_bounds__(256) void pair_kernel(const float* __restrict__ tris,
                                                   const float* __restrict__ planes,
                                                   unsigned* __restrict__ maskW) {
    int b = blockIdx.y;
    // decode upper-triangular tile pair (tf <= tg)
    int t = blockIdx.x;
    int tf = 0, rem = t;
    while (rem >= TILES - tf) { rem -= TILES - tf; ++tf; }
    int tg = tf + rem;

    __shared__ float sFpl[16][4];     // planes of f-tile (Nx,Ny,Nz,d)
    __shared__ float sGpl[16][4];
    __shared__ float sFv[16][3][3];   // vertices of f-tile
    __shared__ float sGv[16][3][3];
    __shared__ float sDu[3][16][16];  // Du_k[m=f][n=g]
    __shared__ float sDv[3][16][16];  // Dv_k[m=g][n=f]
    __shared__ int   sBits[16][16];

    const int tid = threadIdx.x;

    const float* plF = planes + ((size_t)b * TRI_F + tf * 16) * 4;
    const float* plG = planes + ((size_t)b * TRI_F + tg * 16) * 4;
    const float* vtF = tris   + ((size_t)b * TRI_F + tf * 16) * 9;
    const float* vtG = tris   + ((size_t)b * TRI_F + tg * 16) * 9;

#if USE_ASYNC_LDS
    {
        if (tid < 64) {
            __builtin_amdgcn_global_load_async_to_lds_b32(
                (int*)(const_cast<float*>(plF) + tid), (int*)&((float*)sFpl)[tid], 0, 0);
            __builtin_amdgcn_global_load_async_to_lds_b32(
                (int*)(const_cast<float*>(plG) + tid), (int*)&((float*)sGpl)[tid], 0, 0);
        }
        if (tid < 144) {
            __builtin_amdgcn_global_load_async_to_lds_b32(
                (int*)(const_cast<float*>(vtF) + tid), (int*)&((float*)sFv)[tid], 0, 0);
            __builtin_amdgcn_global_load_async_to_lds_b32(
                (int*)(const_cast<float*>(vtG) + tid), (int*)&((float*)sGv)[tid], 0, 0);
        }
        asm volatile("s_wait_asynccnt 0x0" ::: "memory");
    }
#else
    if (tid < 64) {
        ((float*)sFpl)[tid] = plF[tid];
        ((float*)sGpl)[tid] = plG[tid];
    }
    if (tid < 144) {
        ((float*)sFv)[tid] = vtF[tid];
        ((float*)sGv)[tid] = vtG[tid];
    }
#endif
    __syncthreads();

    // --- WMMA stage: 6 waves, one v_wmma_f32_16x16x4_f32 each -------------
    int wave = tid >> 5;
    int lane = tid & 31;
    if (wave < 6) {
        const int  k    = wave % 3;      // which vertex of the "other" triangle
        const bool doDu = (wave < 3);    // Du: A = f-planes, B = g-verts
        const float (*pl)[4]    = doDu ? sFpl : sGpl;
        const float (*vv)[3][3] = doDu ? sGv  : sFv;

        const int hi  = lane >> 4;       // lane half selects K block {0,1} vs {2,3}
        const int l16 = lane & 15;
        const int K0  = hi * 2;

        // A: 16x4 f32. lanes0-15: M=lane, VGPR0=K0,VGPR1=K1; lanes16-31: K2,K3
        v2f A;
        A.x = pl[l16][K0 + 0];
        A.y = pl[l16][K0 + 1];
        // B: 4x16 f32, column N = triangle, rows = (x,y,z,1)
        v2f Bm;
        Bm.x = (K0 + 0 < 3) ? vv[l16][k][K0 + 0] : 1.0f;
        Bm.y = (K0 + 1 < 3) ? vv[l16][k][K0 + 1] : 1.0f;

        v8f C = {0.f, 0.f, 0.f, 0.f, 0.f, 0.f, 0.f, 0.f};
        C = __builtin_amdgcn_wmma_f32_16x16x4_f32(
                /*neg_a=*/false, A, /*neg_b=*/false, Bm,
                /*c_mod=*/(short)0, C, /*reuse_a=*/false, /*reuse_b=*/false);

        float* dst = doDu ? &sDu[k][0][0] : &sDv[k][0][0];
#pragma unroll
        for (int r = 0; r < 8; ++r)
            dst[(r + hi * 8) * 16 + l16] = C[r];   // C layout: VGPR r -> M=r / r+8
    }
    __syncthreads();

    // --- Scalar Moller stage: one pair per thread -------------------------
    const int fi = tid >> 4, gi = tid & 15;
    const int f = tf * 16 + fi, g = tg * 16 + gi;
    bool collide = false;
    if (f < g) {
        float du0 = sDu[0][fi][gi], du1 = sDu[1][fi][gi], du2 = sDu[2][fi][gi];
        float dv0 = sDv[0][gi][fi], dv1 = sDv[1][gi][fi], dv2 = sDv[2][gi][fi];
        if (fabsf(du0) < EPSF) du0 = 0.0f;
        if (fabsf(du1) < EPSF) du1 = 0.0f;
        if (fabsf(du2) < EPSF) du2 = 0.0f;
        if (fabsf(dv0) < EPSF) dv0 = 0.0f;
        if (fabsf(dv1) < EPSF) dv1 = 0.0f;
        if (fabsf(dv2) < EPSF) dv2 = 0.0f;

        bool no_u = ((du0 * du1) > 0.0f) && ((du0 * du2) > 0.0f);
        bool no_v = ((dv0 * dv1) > 0.0f) && ((dv0 * dv2) > 0.0f);
        bool copl = (du0 == 0.0f) && (du1 == 0.0f) && (du2 == 0.0f);

        // intersection line direction D = cross(Nf, Ng); axis = argmax |D|
        float nfx = sFpl[fi][0], nfy = sFpl[fi][1], nfz = sFpl[fi][2];
        float ngx = sGpl[gi][0], ngy = sGpl[gi][1], ngz = sGpl[gi][2];
        float D0 = nfy * ngz - nfz * ngy;
        float D1 = nfz * ngx - nfx * ngz;
        float D2 = nfx * ngy - nfy * ngx;
        float a0 = fabsf(D0), a1 = fabsf(D1), a2 = fabsf(D2);
        int axis = 0; float best = a0;
        if (a1 > best) { axis = 1; best = a1; }
        if (a2 > best) { axis = 2; best = a2; }

        float vp0 = sFv[fi][0][axis], vp1 = sFv[fi][1][axis], vp2 = sFv[fi][2][axis];
        float up0 = sGv[gi][0][axis], up1 = sGv[gi][1][axis], up2 = sGv[gi][2][axis];

        float lo1, hi1, lo2, hi2;
        interval(vp0, vp1, vp2, dv0, dv1, dv2, lo1, hi1);   // triangle f vs plane g
        interval(up0, up1, up2, du0, du1, du2, lo2, hi2);   // triangle g vs plane f
        bool overlap = fmaxf(lo1, lo2) <= fminf(hi1, hi2);

        // shared-vertex exclusion (exact float equality, any (k,l))
        bool shared = false;
#pragma unroll
        for (int kk = 0; kk < 3; ++kk)
#pragma unroll
            for (int ll = 0; ll < 3; ++ll)
                shared = shared || (sFv[fi][kk][0] == sGv[gi][ll][0] &&
                                    sFv[fi][kk][1] == sGv[gi][ll][1] &&
                                    sFv[fi][kk][2] == sGv[gi][ll][2]);

        collide = (!no_u) && (!no_v) && (!copl) && overlap && (!shared);
    }
    sBits[fi][gi] = collide ? 1 : 0;
    __syncthreads();

    // --- Pack 16-bit row masks; word index = f * TILES + tg (flat order) --
    if (tid < 16) {
        unsigned m = 0;
#pragma unroll
        for (int j = 0; j < 16; ++j)
            m |= (sBits[tid][j] ? 1u : 0u) << j;
        maskW[(size_t)b * WPB + (size_t)(tf * 16 + tid) * TILES + tg] = m;
    }
}

// ---------------------------------------------------------------------------
// Compaction: chunk popcounts -> per-batch scan -> ordered scatter
// ---------------------------------------------------------------------------
__global__ __launch_bounds__(256) void chunk_sum_kernel(const unsigned* __restrict__ maskW,
                                                        unsigned* __restrict__ sums) {
    int b = blockIdx.y, c = blockIdx.x, tid = threadIdx.x;
    unsigned m = maskW[(size_t)b * WPB + c * 256 + tid] & 0xFFFFu;
    __shared__ unsigned s[256];
    s[tid] = (unsigned)__popc(m);
    __syncthreads();
    for (int off = 128; off > 0; off >>= 1) {
        if (tid < off) s[tid] += s[tid + off];
        __syncthreads();
    }
    if (tid == 0) sums[b * 256 + c] = s[0];
}

__global__ __launch_bounds__(256) void scan_kernel(const unsigned* __restrict__ sums,
                                                   unsigned* __restrict__ bases) {
    int b = blockIdx.x, tid = threadIdx.x;
    __shared__ unsigned bufA[256], bufB[256];
    bufA[tid] = sums[b * 256 + tid];
    __syncthreads();
    unsigned* in  = bufA;
    unsigned* out = bufB;
    for (int off = 1; off < 256; off <<= 1) {
        out[tid] = in[tid] + ((tid >= off) ? in[tid - off] : 0u);
        __syncthreads();
        unsigned* tmp = in; in = out; out = tmp;
    }
    bases[b * 256 + tid] = (tid > 0) ? in[tid - 1] : 0u;   // exclusive scan
}

__global__ __launch_bounds__(256) void scatter_kernel(const unsigned* __restrict__ maskW,
                                                      const unsigned* __restrict__ bases,
                                                      int* __restrict__ out) {
    int b = blockIdx.y, c = blockIdx.x, tid = threadIdx.x;
    int word = c * 256 + tid;
    unsigned m = maskW[(size_t)b * WPB + word] & 0xFFFFu;
    __shared__ unsigned bufA[256], bufB[256];
    unsigned pc = (unsigned)__popc(m);
    bufA[tid] = pc;
    __syncthreads();
    unsigned* in   = bufA;
    unsigned* outp = bufB;
    for (int off = 1; off < 256; off <<= 1) {
        outp[tid] = in[tid] + ((tid >= off) ? in[tid - off] : 0u);
        __syncthreads();
        unsigned* tmp = in; in = outp; outp = tmp;
    }
    unsigned slot = bases[b * 256 + c] + in[tid] - pc;     // exclusive within chunk
    int f  = word >> 6;        // word / TILES
    int tg = word & 63;        // word % TILES
    unsigned mm = m;
    while (mm) {
        int bit = __ffs(mm) - 1;
        mm &= mm - 1;
        if (slot < KOUT) {
            int g = tg * 16 + bit;
            out[((size_t)b * KOUT + slot) * 2 + 0] = f;
            out[((size_t)b * KOUT + slot) * 2 + 1] = g;
        }
        ++slot;
    }
}

// ---------------------------------------------------------------------------
extern "C" void kernel_launch(void* const* d_in, const int* in_sizes, int n_in,
                              void* d_out, int out_size, void* d_ws, size_t ws_size,
                              hipStream_t stream) {
    const float* tris = (const float*)d_in[0];
    int* out = (int*)d_out;
    int Bn = in_sizes[0] / (TRI_F * 9);   // = 2 for the reference inputs

    char* ws = (char*)d_ws;
    float* planes      = (float*)ws;
    size_t planesBytes = (size_t)Bn * TRI_F * 4 * sizeof(float);
    unsigned* maskW    = (unsigned*)(ws + planesBytes);
    size_t maskBytes   = (size_t)Bn * WPB * sizeof(unsigned);
    unsigned* sums     = (unsigned*)(ws + planesBytes + maskBytes);
    size_t sumsBytes   = (size_t)Bn * 256 * sizeof(unsigned);
    unsigned* bases    = (unsigned*)(ws + planesBytes + maskBytes + sumsBytes);

    // output padded with -1; mask words in the never-visited (tg < tf) region = 0
    (void)hipMemsetAsync(out, 0xFF, (size_t)out_size * sizeof(int), stream);
    (void)hipMemsetAsync(maskW, 0, maskBytes, stream);

    int nTri = Bn * TRI_F;
    plane_kernel<<<dim3((nTri + 255) / 256), 256, 0, stream>>>(tris, planes, nTri);
    pair_kernel<<<dim3(NTP, Bn), 256, 0, stream>>>(tris, planes, maskW);
    chunk_sum_kernel<<<dim3(256, Bn), 256, 0, stream>>>(maskW, sums);
    scan_kernel<<<dim3(Bn), 256, 0, stream>>>(sums, bases);
    scatter_kernel<<<dim3(256, Bn), 256, 0, stream>>>(maskW, bases, out);
}